// GATN_81432579932953
// MI455X (gfx1250) — compile-verified
//
#include <hip/hip_runtime.h>
#include <cstring>
#include <cstddef>

// ---------------- problem constants (match reference) ----------------
constexpr int B_   = 512;
constexpr int NPG  = 128;
constexpr int DEG  = 16;
constexpr int N_   = B_ * NPG;        // 65536 nodes
constexpr int E_   = N_ * DEG;        // 1048576 edges
constexpr int IN_DIM = 11, HID = 32, HEADS = 4;
constexpr int F    = HEADS * HID;     // 128
constexpr int K1   = 103;             // ceil(0.8*128)
constexpr int K2   = 83;              // ceil(0.8*103)
constexpr int N2   = B_ * K1;         // 52736 pooled nodes (multiple of 16)

typedef float v2f __attribute__((ext_vector_type(2)));
typedef float v8f __attribute__((ext_vector_type(8)));

// ---------------- order-preserving float<->uint for atomicMax --------
__device__ __forceinline__ unsigned f2ord(float f) {
    unsigned u = __float_as_uint(f);
    return (u & 0x80000000u) ? ~u : (u | 0x80000000u);
}
__device__ __forceinline__ float ord2f(unsigned u) {
    return (u & 0x80000000u) ? __uint_as_float(u & 0x7FFFFFFFu)
                             : __uint_as_float(~u);
}
static unsigned h_f2ord(float f) {
    unsigned u; memcpy(&u, &f, 4);
    return (u & 0x80000000u) ? ~u : (u | 0x80000000u);
}

// =====================================================================
// Generic f32 WMMA GEMM: C[M,Nn] = A[M,K] * B[K,Nn] (+bias, opt ReLU)
// One wave per 16x16 C tile, V_WMMA_F32_16X16X4_F32 along K.
// Main K loop is guard-free (all loads in-bounds); the K%4 tail chunk
// (only K=11 here) uses clamped addresses + v_cndmask selects so there
// is never EXEC divergence around loads or WMMA.
// Requires M % 16 == 0 and Nn % 16 == 0 (true for every call here).
// =====================================================================
__global__ void k_gemm_wmma(const float* __restrict__ A,
                            const float* __restrict__ Bm,
                            float* __restrict__ C,
                            const float* __restrict__ bias,
                            int M, int Nn, int K, int relu)
{
    const int wavesPerBlk = blockDim.x >> 5;
    const int  ntn    = Nn >> 4;
    const long ntiles = (long)(M >> 4) * ntn;
    const long tile   = (long)blockIdx.x * wavesPerBlk + (threadIdx.x >> 5);
    if (tile >= ntiles) return;               // wave-uniform -> EXEC stays full

    const int lane = threadIdx.x & 31;
    const int lr   = lane & 15;               // row (A) / col (B,C) within tile
    const int half = lane >> 4;               // K sub-select
    const int tm   = (int)(tile / ntn);
    const int tn   = (int)(tile % ntn);

    const float* Arow = A  + (size_t)(tm * 16 + lr) * K;
    const float* Bcol = Bm + tn * 16 + lr;

    v8f acc = {};

    const int Kmain = K & ~3;                 // full 4-wide chunks
    for (int kc = 0; kc < Kmain; kc += 4) {
        const int k0 = kc + 2 * half;         // < K, always in-bounds
        v2f a, b;
        a.x = Arow[k0];
        a.y = Arow[k0 + 1];
        b.x = Bcol[(size_t)k0 * Nn];
        b.y = Bcol[(size_t)(k0 + 1) * Nn];
        acc = __builtin_amdgcn_wmma_f32_16x16x4_f32(
                  false, a, false, b, (short)0, acc, false, false);
    }
    if (Kmain < K) {                          // remainder chunk (K=11 case)
        const int k0 = Kmain + 2 * half;
        const int k1 = k0 + 1;
        const int c0 = (k0 < K) ? k0 : (K - 1);   // clamped, in-bounds
        const int c1 = (k1 < K) ? k1 : (K - 1);
        const float a0 = Arow[c0], a1 = Arow[c1];
        const float b0 = Bcol[(size_t)c0 * Nn], b1 = Bcol[(size_t)c1 * Nn];
        v2f a, b;
        a.x = (k0 < K) ? a0 : 0.f;            // v_cndmask, no EXEC change
        a.y = (k1 < K) ? a1 : 0.f;
        b.x = (k0 < K) ? b0 : 0.f;
        b.y = (k1 < K) ? b1 : 0.f;
        acc = __builtin_amdgcn_wmma_f32_16x16x4_f32(
                  false, a, false, b, (short)0, acc, false, false);
    }

    const int col = tn * 16 + lr;
    const float bv = bias ? bias[col] : 0.f;
#pragma unroll
    for (int r = 0; r < 8; ++r) {
        const int row = tm * 16 + r + 8 * half;
        float v = acc[r] + bv;
        if (relu) v = fmaxf(v, 0.f);
        C[(size_t)row * Nn + col] = v;
    }
}

// ---------------- attention coefficients: as_[n,h], ad_[n,h] ---------
__global__ void k_attn_coef(const float* __restrict__ xp,
                            const float* __restrict__ a_s,
                            const float* __restrict__ a_d,
                            float* __restrict__ as_, float* __restrict__ ad_,
                            int nNodes)
{
    const int i = blockIdx.x * blockDim.x + threadIdx.x;   // node*HEADS + h
    if (i >= nNodes * HEADS) return;
    const int n = i >> 2, h = i & 3;
    const float* xr  = xp  + (size_t)n * F + h * HID;
    const float* asr = a_s + h * HID;
    const float* adr = a_d + h * HID;
    float s = 0.f, d = 0.f;
#pragma unroll
    for (int c = 0; c < HID; ++c) { float v = xr[c]; s += v * asr[c]; d += v * adr[c]; }
    as_[i] = s; ad_[i] = d;
}

// ---------------- fills ----------------------------------------------
__global__ void k_fill_f32(float* p, float v, long n) {
    long i = (long)blockIdx.x * blockDim.x + threadIdx.x;
    if (i < n) p[i] = v;
}
__global__ void k_fill_u32(unsigned* p, unsigned v, long n) {
    long i = (long)blockIdx.x * blockDim.x + threadIdx.x;
    if (i < n) p[i] = v;
}

// ---------------- segment softmax: pass 1 (max) ----------------------
// Iterates real edges [0,nEdges) plus self-loops [nEdges, nEdges+nNodes).
__global__ void k_edge_max(const int* __restrict__ src, const int* __restrict__ dst,
                           const int* __restrict__ msk,
                           const float* __restrict__ as_, const float* __restrict__ ad_,
                           unsigned* __restrict__ mx, int nEdges, int nNodes)
{
    const int e = blockIdx.x * blockDim.x + threadIdx.x;
    if (e >= nEdges + nNodes) return;
    int s, d;
    if (e < nEdges) { if (msk && !msk[e]) return; s = src[e]; d = dst[e]; }
    else            { s = d = e - nEdges; }
#pragma unroll
    for (int h = 0; h < HEADS; ++h) {
        float l = as_[s * HEADS + h] + ad_[d * HEADS + h];
        l = (l >= 0.f) ? l : 0.2f * l;                     // leaky_relu(0.2)
        atomicMax(&mx[d * HEADS + h], f2ord(l));
    }
}

// ---------------- segment softmax: pass 2 (denominator) --------------
__global__ void k_edge_den(const int* __restrict__ src, const int* __restrict__ dst,
                           const int* __restrict__ msk,
                           const float* __restrict__ as_, const float* __restrict__ ad_,
                           const unsigned* __restrict__ mx, float* __restrict__ den,
                           int nEdges, int nNodes)
{
    const int e = blockIdx.x * blockDim.x + threadIdx.x;
    if (e >= nEdges + nNodes) return;
    int s, d;
    if (e < nEdges) { if (msk && !msk[e]) return; s = src[e]; d = dst[e]; }
    else            { s = d = e - nEdges; }
#pragma unroll
    for (int h = 0; h < HEADS; ++h) {
        float l = as_[s * HEADS + h] + ad_[d * HEADS + h];
        l = (l >= 0.f) ? l : 0.2f * l;
        atomicAdd(&den[d * HEADS + h], expf(l - ord2f(mx[d * HEADS + h])));
    }
}

// ---------------- segment softmax: pass 3 (weighted aggregate) -------
// One thread per (edge-or-loop, head); 32 scatter-adds each.
__global__ void k_edge_agg(const int* __restrict__ src, const int* __restrict__ dst,
                           const int* __restrict__ msk,
                           const float* __restrict__ as_, const float* __restrict__ ad_,
                           const unsigned* __restrict__ mx, const float* __restrict__ den,
                           const float* __restrict__ xp, float* __restrict__ agg,
                           int nEdges, int nNodes)
{
    const int i = blockIdx.x * blockDim.x + threadIdx.x;
    if (i >= (nEdges + nNodes) * HEADS) return;
    const int e = i >> 2, h = i & 3;
    int s, d;
    if (e < nEdges) { if (msk && !msk[e]) return; s = src[e]; d = dst[e]; }
    else            { s = d = e - nEdges; }
    float l = as_[s * HEADS + h] + ad_[d * HEADS + h];
    l = (l >= 0.f) ? l : 0.2f * l;
    const float ex    = expf(l - ord2f(mx[d * HEADS + h]));
    const float alpha = ex / (den[d * HEADS + h] + 1e-16f);
    const float* xr = xp  + (size_t)s * F + h * HID;
    float*       og = agg + (size_t)d * F + h * HID;
#pragma unroll
    for (int c = 0; c < HID; ++c) atomicAdd(&og[c], alpha * xr[c]);
}

// ---------------- bias + relu in place (ld = 128) --------------------
__global__ void k_bias_relu(float* __restrict__ x, const float* __restrict__ b, long n)
{
    long i = (long)blockIdx.x * blockDim.x + threadIdx.x;
    if (i < n) x[i] = fmaxf(x[i] + b[i & (F - 1)], 0.f);
}

// ---------------- TopK pooling (exact, rank by count) ----------------
// One block per graph (128 threads). score = tanh((h.w)/||w||);
// rank r < k kept, new index = b*k + rank (matches jax top_k ordering
// up to tie-breaks, which cannot change the kept set's contribution).
__global__ void k_topk_pool(const float* __restrict__ h, const float* __restrict__ w,
                            float* __restrict__ outX, int* __restrict__ remap,
                            int npg, int k)
{
    __shared__ float sc[NPG];
    const int b = blockIdx.x, tid = threadIdx.x;

    float nrm2 = 0.f;
#pragma unroll
    for (int c = 0; c < HID; ++c) nrm2 += w[c] * w[c];
    const float inv = 1.f / sqrtf(nrm2);

    const int node = b * npg + tid;
    float mysc = 0.f;
    if (tid < npg) {
        const float* hr = h + (size_t)node * HID;
        float dp = 0.f;
#pragma unroll
        for (int c = 0; c < HID; ++c) dp += hr[c] * w[c];
        mysc = tanhf(dp * inv);
        sc[tid] = mysc;
    }
    __syncthreads();
    if (tid < npg) {
        int rank = 0;
        for (int j = 0; j < npg; ++j) {
            const float sj = sc[j];
            rank += (sj > mysc) || (sj == mysc && j < tid);
        }
        if (remap) remap[node] = (rank < k) ? (b * k + rank) : -1;
        if (rank < k) {
            const float* hr = h + (size_t)node * HID;
            float* o = outX + (size_t)(b * k + rank) * HID;
#pragma unroll
            for (int c = 0; c < HID; ++c) o[c] = hr[c] * mysc;
        }
    }
}

// ---------------- edge remap after pool ------------------------------
__global__ void k_remap_edges(const int* __restrict__ src, const int* __restrict__ dst,
                              const int* __restrict__ remap,
                              int* __restrict__ src2, int* __restrict__ dst2,
                              int* __restrict__ msk2, int nEdges)
{
    const int e = blockIdx.x * blockDim.x + threadIdx.x;
    if (e >= nEdges) return;
    const int ns = remap[src[e]], nd = remap[dst[e]];
    const int m = (ns >= 0) && (nd >= 0);
    src2[e] = m ? ns : 0; dst2[e] = m ? nd : 0; msk2[e] = m;
}

// ---------------- per-graph readout: [max | mean] --------------------
__global__ void k_readout(const float* __restrict__ hp, float* __restrict__ g, int k)
{
    const int b = blockIdx.x, c = threadIdx.x;       // blockDim = 32
    const float* base = hp + (size_t)b * k * HID + c;
    float mx = -1e30f, sm = 0.f;
    for (int r = 0; r < k; ++r) { float v = base[(size_t)r * HID]; mx = fmaxf(mx, v); sm += v; }
    g[b * 2 * HID + c]       = mx;
    g[b * 2 * HID + HID + c] = sm / (float)k;
}

__global__ void k_add(const float* a, const float* b, float* o, int n)
{
    const int i = blockIdx.x * blockDim.x + threadIdx.x;
    if (i < n) o[i] = a[i] + b[i];
}

// ---------------- final [512,1024] @ [1024,1] + b --------------------
__global__ void k_mlp3(const float* __restrict__ X, const float* __restrict__ W,
                       const float* __restrict__ bias, float* __restrict__ out)
{
    __shared__ float red[256];
    const int b = blockIdx.x, tid = threadIdx.x;
    const float* xr = X + (size_t)b * 1024;
    float s = 0.f;
    for (int i = tid; i < 1024; i += 256) s += xr[i] * W[i];
    red[tid] = s; __syncthreads();
    for (int st = 128; st > 0; st >>= 1) { if (tid < st) red[tid] += red[tid + st]; __syncthreads(); }
    if (tid == 0) out[b] = red[0] + bias[0];
}

// =====================================================================
static inline int nblk(long n, int t) { return (int)((n + t - 1) / t); }

extern "C" void kernel_launch(void* const* d_in, const int* in_sizes, int n_in,
                              void* d_out, int out_size, void* d_ws, size_t ws_size,
                              hipStream_t stream)
{
    (void)in_sizes; (void)n_in; (void)out_size; (void)ws_size;

    // ---- inputs (setup_inputs order) ----
    const float* x        = (const float*)d_in[0];
    const int*   ei       = (const int*)  d_in[1];
    const int*   src      = ei;
    const int*   dst      = ei + E_;
    const float* W_g1     = (const float*)d_in[4];
    const float* att_src1 = (const float*)d_in[5];
    const float* att_dst1 = (const float*)d_in[6];
    const float* b_g1     = (const float*)d_in[7];
    const float* W_t1     = (const float*)d_in[8];
    const float* b_t1     = (const float*)d_in[9];
    const float* pool_w1  = (const float*)d_in[10];
    const float* W_g2     = (const float*)d_in[11];
    const float* att_src2 = (const float*)d_in[12];
    const float* att_dst2 = (const float*)d_in[13];
    const float* b_g2     = (const float*)d_in[14];
    const float* W_t2     = (const float*)d_in[15];
    const float* b_t2     = (const float*)d_in[16];
    const float* pool_w2  = (const float*)d_in[17];
    const float* W_l1     = (const float*)d_in[18];
    const float* b_l1     = (const float*)d_in[19];
    const float* W_l2     = (const float*)d_in[20];
    const float* b_l2     = (const float*)d_in[21];
    const float* W_l3     = (const float*)d_in[22];
    const float* b_l3     = (const float*)d_in[23];
    float* out = (float*)d_out;

    // ---- workspace bump allocator (layer-2 reuses layer-1 buffers) ----
    char* ws = (char*)d_ws; size_t off = 0;
    auto alloc = [&](size_t bytes) -> void* {
        void* p = ws + off; off = (off + bytes + 255) & ~(size_t)255; return p;
    };
    float*    xp    = (float*)   alloc((size_t)N_ * F * 4);     // xp1 / xp2
    float*    agg   = (float*)   alloc((size_t)N_ * F * 4);     // agg1 / agg2
    float*    asb   = (float*)   alloc((size_t)N_ * HEADS * 4);
    float*    adb   = (float*)   alloc((size_t)N_ * HEADS * 4);
    unsigned* mxb   = (unsigned*)alloc((size_t)N_ * HEADS * 4);
    float*    denb  = (float*)   alloc((size_t)N_ * HEADS * 4);
    float*    h1    = (float*)   alloc((size_t)N_ * HID * 4);   // h1 / h2
    int*      remap = (int*)     alloc((size_t)N_ * 4);
    float*    hp1   = (float*)   alloc((size_t)N2 * HID * 4);
    int*      src2  = (int*)     alloc((size_t)E_ * 4);
    int*      dst2  = (int*)     alloc((size_t)E_ * 4);
    int*      msk2  = (int*)     alloc((size_t)E_ * 4);
    float*    g1    = (float*)   alloc((size_t)B_ * 64 * 4);
    float*    hp2   = (float*)   alloc((size_t)B_ * K2 * HID * 4);
    float*    g2    = (float*)   alloc((size_t)B_ * 64 * 4);
    float*    gsum  = (float*)   alloc((size_t)B_ * 64 * 4);
    float*    m1    = (float*)   alloc((size_t)B_ * 256 * 4);
    float*    m2    = (float*)   alloc((size_t)B_ * 1024 * 4);

    const unsigned ORD_NEG = h_f2ord(-1e30f);
    const int T = 256;
    auto gemm = [&](const float* A, const float* Bm, float* C, const float* bias,
                    int M, int Nn, int K, int relu) {
        long tiles = (long)(M / 16) * (Nn / 16);
        k_gemm_wmma<<<nblk(tiles, 8), 256, 0, stream>>>(A, Bm, C, bias, M, Nn, K, relu);
    };

    // ================= GAT layer 1 (N_ nodes, E_ edges, all valid) =====
    gemm(x, W_g1, xp, nullptr, N_, F, IN_DIM, 0);                         // xp1 = x @ W_g1
    k_attn_coef<<<nblk((long)N_ * HEADS, T), T, 0, stream>>>(xp, att_src1, att_dst1, asb, adb, N_);
    k_fill_f32<<<nblk((long)N_ * F, T), T, 0, stream>>>(agg, 0.f, (long)N_ * F);
    k_fill_u32<<<nblk((long)N_ * HEADS, T), T, 0, stream>>>(mxb, ORD_NEG, (long)N_ * HEADS);
    k_fill_f32<<<nblk((long)N_ * HEADS, T), T, 0, stream>>>(denb, 0.f, (long)N_ * HEADS);
    k_edge_max<<<nblk((long)E_ + N_, T), T, 0, stream>>>(src, dst, nullptr, asb, adb, mxb, E_, N_);
    k_edge_den<<<nblk((long)E_ + N_, T), T, 0, stream>>>(src, dst, nullptr, asb, adb, mxb, denb, E_, N_);
    k_edge_agg<<<nblk(((long)E_ + N_) * HEADS, T), T, 0, stream>>>(src, dst, nullptr, asb, adb, mxb, denb, xp, agg, E_, N_);
    k_bias_relu<<<nblk((long)N_ * F, T), T, 0, stream>>>(agg, b_g1, (long)N_ * F);
    gemm(agg, W_t1, h1, b_t1, N_, HID, F, 0);                             // h1 = relu(gat1) @ W_t1 + b_t1

    // ================= pool 1 + readout 1 ==============================
    k_topk_pool<<<B_, 128, 0, stream>>>(h1, pool_w1, hp1, remap, NPG, K1);
    k_remap_edges<<<nblk(E_, T), T, 0, stream>>>(src, dst, remap, src2, dst2, msk2, E_);
    k_readout<<<B_, 32, 0, stream>>>(hp1, g1, K1);

    // ================= GAT layer 2 (N2 nodes, masked edges) ============
    gemm(hp1, W_g2, xp, nullptr, N2, F, HID, 0);                          // xp2 = hp1 @ W_g2
    k_attn_coef<<<nblk((long)N2 * HEADS, T), T, 0, stream>>>(xp, att_src2, att_dst2, asb, adb, N2);
    k_fill_f32<<<nblk((long)N2 * F, T), T, 0, stream>>>(agg, 0.f, (long)N2 * F);
    k_fill_u32<<<nblk((long)N2 * HEADS, T), T, 0, stream>>>(mxb, ORD_NEG, (long)N2 * HEADS);
    k_fill_f32<<<nblk((long)N2 * HEADS, T), T, 0, stream>>>(denb, 0.f, (long)N2 * HEADS);
    k_edge_max<<<nblk((long)E_ + N2, T), T, 0, stream>>>(src2, dst2, msk2, asb, adb, mxb, E_, N2);
    k_edge_den<<<nblk((long)E_ + N2, T), T, 0, stream>>>(src2, dst2, msk2, asb, adb, mxb, denb, E_, N2);
    k_edge_agg<<<nblk(((long)E_ + N2) * HEADS, T), T, 0, stream>>>(src2, dst2, msk2, asb, adb, mxb, denb, xp, agg, E_, N2);
    k_bias_relu<<<nblk((long)N2 * F, T), T, 0, stream>>>(agg, b_g2, (long)N2 * F);
    gemm(agg, W_t2, h1, b_t2, N2, HID, F, 0);                             // h2 (reuses h1 buffer)

    // ================= pool 2 + readout 2 ==============================
    k_topk_pool<<<B_, 128, 0, stream>>>(h1, pool_w2, hp2, nullptr, K1, K2);
    k_readout<<<B_, 32, 0, stream>>>(hp2, g2, K2);
    k_add<<<nblk((long)B_ * 64, T), T, 0, stream>>>(g1, g2, gsum, B_ * 64);

    // ================= MLP head ========================================
    gemm(gsum, W_l1, m1, b_l1, B_, 256, 64, 1);
    gemm(m1,   W_l2, m2, b_l2, B_, 1024, 256, 1);
    k_mlp3<<<B_, 256, 0, stream>>>(m2, W_l3, b_l3, out);
}